// RoadSignNetLoss_34754875359735
// MI455X (gfx1250) — compile-verified
//
#include <hip/hip_runtime.h>
#include <math.h>

#define NCLS 3
#define NANC 3
#define MBOX 50
#define BATCH 128

// ---------------- scale constants (compile-time via templates) ----------------
template<int S> __device__ __forceinline__ float STRv() { return S == 0 ? 8.f : (S == 1 ? 16.f : 32.f); }
template<int S> __device__ __forceinline__ int   Hv()   { return S == 0 ? 80 : (S == 1 ? 40 : 20); }
template<int S> __device__ __forceinline__ float AWv(int k) {
    if (S == 0) return k == 0 ? 10.f : (k == 1 ? 16.f : 33.f);
    if (S == 1) return k == 0 ? 30.f : (k == 1 ? 62.f : 59.f);
    return k == 0 ? 116.f : (k == 1 ? 156.f : 373.f);
}
template<int S> __device__ __forceinline__ float AHv(int k) {
    if (S == 0) return k == 0 ? 13.f : (k == 1 ? 30.f : 23.f);
    if (S == 1) return k == 0 ? 61.f : (k == 1 ? 45.f : 119.f);
    return k == 0 ? 90.f : (k == 1 ? 198.f : 326.f);
}

// ---------------- math helpers ----------------
__device__ __forceinline__ float splus(float x) {           // softplus = bce(x, 0); bce(x,1)=splus(-x)
    return fmaxf(x, 0.f) + log1pf(expf(-fabsf(x)));
}
__device__ __forceinline__ float sl1(float a, float b) {
    float d = fabsf(a - b);
    return d < 1.f ? 0.5f * d * d : d - 0.5f;
}
__device__ __forceinline__ float sigm(float x) { return 1.f / (1.f + expf(-x)); }

// ---------------- WMMA-based wave32 reduction (exact f32 add tree) ----------------
typedef __attribute__((ext_vector_type(2))) float v2f;
typedef __attribute__((ext_vector_type(8))) float v8f;

__device__ __forceinline__ float wave_reduce(float x) {
#if defined(__has_builtin) && __has_builtin(__builtin_amdgcn_wmma_f32_16x16x4_f32)
    // A(16x4): row r = {v_r, 0, v_{r+16}, 0}.  B(4x16) = ones.
    // D[r][j] = v_r + v_{r+16}.  Lane L holds D rows (L<16 ? 0..7 : 8..15) in 8 VGPRs.
    v2f A; A.x = x; A.y = 0.f;
    v2f B; B.x = 1.f; B.y = 1.f;
    v8f C = {};
    v8f D = __builtin_amdgcn_wmma_f32_16x16x4_f32(false, A, false, B, (short)0, C, false, false);
    float p = D[0] + D[1] + D[2] + D[3] + D[4] + D[5] + D[6] + D[7];
    p += __shfl_xor(p, 16, 32);   // rows 0..7 half + rows 8..15 half = full 32-lane sum
    return p;
#else
    for (int off = 16; off > 0; off >>= 1) x += __shfl_xor(x, off, 32);
    return x;
#endif
}

// ---------------- init: zero the 16 accumulators ----------------
__global__ void init_acc_kernel(float* __restrict__ acc) {
    if (threadIdx.x < 16) acc[threadIdx.x] = 0.f;
}

// ---------------- obj baseline: sum 0.5*softplus(x) over all elements ----------------
__global__ void obj_base_kernel(const float* __restrict__ p, int n4, float* __restrict__ slot) {
    float local = 0.f;
    int stride = gridDim.x * blockDim.x;
    for (int i = blockIdx.x * blockDim.x + threadIdx.x; i < n4; i += stride) {
        float4 x = reinterpret_cast<const float4*>(p)[i];
        local += 0.5f * (splus(x.x) + splus(x.y) + splus(x.z) + splus(x.w));
    }
    float tot = wave_reduce(local);
    if ((threadIdx.x & 31) == 0) atomicAdd(slot, tot);
}

// ---------------- per-GT mapping ----------------
template<int S>
__device__ __forceinline__ void map_gt(float cx, float cy, float w, float h,
                                       int& a, int& gi, int& gj) {
    float ws = fmaxf(w, 1e-6f), hs = fmaxf(h, 1e-6f);
    float best = -1.f; a = 0;
#pragma unroll
    for (int k = 0; k < 3; ++k) {
        float aw = AWv<S>(k), ah = AHv<S>(k);
        float r = fminf(ws / aw, aw / ws) * fminf(hs / ah, ah / hs);
        if (r > best) { best = r; a = k; }
    }
    float inv = 1.f / STRv<S>();
    int Wd = Hv<S>();
    gi = min(max((int)(cx * inv), 0), Wd - 1);
    gj = min(max((int)(cy * inv), 0), Wd - 1);
}

// v layout per scale: v[0]=obj_corr v[1]=cls v[2]=xy v[3]=wh v[4]=npos
template<int S>
__device__ __forceinline__ void scale_contrib(const float* __restrict__ cls,
                                              const float* __restrict__ box,
                                              const float* __restrict__ obj,
                                              int b, int a, int gi, int gj, int lab,
                                              bool cellw, bool labw,
                                              float cx, float cy, float w, float h,
                                              float* v) {
    const int H = Hv<S>();
    const int HW = H * H;
    const int cell = gj * H + gi;
    if (cellw) {
        v[4] += 1.f;                                   // unique positive cell
        float o = obj[(b * 3 + a) * HW + cell];
        v[0] += splus(-o) - 0.5f * splus(o);           // bce(o,1)*1 - bce(o,0)*0.5
        float cs = 0.f;
#pragma unroll
        for (int c = 0; c < 3; ++c)                    // cls baseline: all-zero targets at cell
            cs += splus(cls[(b * 9 + a * 3 + c) * HW + cell]);
        v[1] += cs;
        float p0 = box[(b * 12 + a * 4 + 0) * HW + cell];
        float p1 = box[(b * 12 + a * 4 + 1) * HW + cell];
        float p2 = box[(b * 12 + a * 4 + 2) * HW + cell];
        float p3 = box[(b * 12 + a * 4 + 3) * HW + cell];
        float inv = 1.f / STRv<S>();
        float tx = cx * inv - (float)gi;
        float ty = cy * inv - (float)gj;
        float tw = logf(w / AWv<S>(a) + 1e-6f);
        float th = logf(h / AHv<S>(a) + 1e-6f);
        v[2] += sl1(sigm(p0), tx) + sl1(sigm(p1), ty);
        v[3] += sl1(p2, tw) + sl1(p3, th);
    }
    if (labw) {                                        // per-(cell,label) one-hot correction
        float pc = cls[(b * 9 + a * 3 + lab) * HW + cell];
        v[1] += splus(-pc) - splus(pc);
    }
}

// ---------------- GT-driven positives kernel: B*M = 6400 threads ----------------
__global__ void pos_kernel(const float* __restrict__ cls0, const float* __restrict__ box0, const float* __restrict__ obj0,
                           const float* __restrict__ cls1, const float* __restrict__ box1, const float* __restrict__ obj1,
                           const float* __restrict__ cls2, const float* __restrict__ box2, const float* __restrict__ obj2,
                           const float* __restrict__ gtb, const int* __restrict__ gtl,
                           float* __restrict__ acc) {
    int t = blockIdx.x * blockDim.x + threadIdx.x;     // exactly B*M threads launched
    int b = t / MBOX;
    int m = t - b * MBOX;

    float4 g = reinterpret_cast<const float4*>(gtb)[t];
    int lab = gtl[t];
    float cx = 0.5f * (g.x + g.z), cy = 0.5f * (g.y + g.w);
    float w = g.z - g.x, h = g.w - g.y;
    bool valid = (lab >= 0) && (w >= 1.f) && (h >= 1.f);

    float v[16];
#pragma unroll
    for (int q = 0; q < 16; ++q) v[q] = 0.f;

    int a0 = 0, gi0 = 0, gj0 = 0, a1 = 0, gi1 = 0, gj1 = 0, a2 = 0, gi2 = 0, gj2 = 0;
    bool cw0 = true, cw1 = true, cw2 = true, lw0 = true, lw1 = true, lw2 = true;

    if (valid) {
        v[15] = 1.f;                                   // Σ valid (for global has_pos gate)
        map_gt<0>(cx, cy, w, h, a0, gi0, gj0);
        map_gt<1>(cx, cy, w, h, a1, gi1, gj1);
        map_gt<2>(cx, cy, w, h, a2, gi2, gj2);
        // deterministic max-m winner rule for colliding scatters
        for (int mp = m + 1; mp < MBOX; ++mp) {
            float4 gp = reinterpret_cast<const float4*>(gtb)[b * MBOX + mp];
            int lp = gtl[b * MBOX + mp];
            float wp = gp.z - gp.x, hp = gp.w - gp.y;
            if (!((lp >= 0) && (wp >= 1.f) && (hp >= 1.f))) continue;
            float cxp = 0.5f * (gp.x + gp.z), cyp = 0.5f * (gp.y + gp.w);
            int ap, gip, gjp;
            map_gt<0>(cxp, cyp, wp, hp, ap, gip, gjp);
            if (ap == a0 && gip == gi0 && gjp == gj0) { cw0 = false; if (lp == lab) lw0 = false; }
            map_gt<1>(cxp, cyp, wp, hp, ap, gip, gjp);
            if (ap == a1 && gip == gi1 && gjp == gj1) { cw1 = false; if (lp == lab) lw1 = false; }
            map_gt<2>(cxp, cyp, wp, hp, ap, gip, gjp);
            if (ap == a2 && gip == gi2 && gjp == gj2) { cw2 = false; if (lp == lab) lw2 = false; }
        }
        scale_contrib<0>(cls0, box0, obj0, b, a0, gi0, gj0, lab, cw0, lw0, cx, cy, w, h, &v[0]);
        scale_contrib<1>(cls1, box1, obj1, b, a1, gi1, gj1, lab, cw1, lw1, cx, cy, w, h, &v[5]);
        scale_contrib<2>(cls2, box2, obj2, b, a2, gi2, gj2, lab, cw2, lw2, cx, cy, w, h, &v[10]);
    }

    // full-EXEC point: WMMA wave reduction of the 16 accumulators
#pragma unroll
    for (int q = 0; q < 16; ++q) {
        float tot = wave_reduce(v[q]);
        if ((threadIdx.x & 31) == 0) atomicAdd(&acc[q], tot);
    }
}

// ---------------- finalize: reference gating & weights ----------------
__global__ void finalize_kernel(const float* __restrict__ acc, float* __restrict__ out) {
    if (threadIdx.x != 0 || blockIdx.x != 0) return;
    const float Ns0 = 128.f * 3.f * 80.f * 80.f;   // 2,457,600
    const float Ns1 = 128.f * 3.f * 40.f * 40.f;   //   614,400
    const float Ns2 = 128.f * 3.f * 20.f * 20.f;   //   153,600
    float tobj = acc[0] / Ns0 + acc[5] / Ns1 + acc[10] / Ns2;
    float tcls = 0.f, tbox = 0.f;
#pragma unroll
    for (int s = 0; s < 3; ++s) {
        float np = acc[s * 5 + 4];
        if (np > 0.f) {                                 // per-scale has = n_pos > 0
            tcls += acc[s * 5 + 1] / fmaxf(np * 3.f, 1.f);
            tbox += (acc[s * 5 + 2] + acc[s * 5 + 3]) / fmaxf(np * 2.f, 1.f);
        }
    }
    tobj *= (1.f / 3.f);
    bool has = acc[15] > 0.f;                           // num_pos_total (= 3*Σvalid) > 0
    float tc = has ? tcls * (1.f / 3.f) : 0.f;
    float tb = has ? tbox * (1.f / 3.f) : 0.f;
    out[0] = 1.f * tc + 5.f * tb + 1.f * tobj;          // L_CLS, L_BOX, L_OBJ
}

// ---------------- host launch ----------------
extern "C" void kernel_launch(void* const* d_in, const int* in_sizes, int n_in,
                              void* d_out, int out_size, void* d_ws, size_t ws_size,
                              hipStream_t stream) {
    const float* cls0 = (const float*)d_in[0];
    const float* box0 = (const float*)d_in[1];
    const float* obj0 = (const float*)d_in[2];
    const float* cls1 = (const float*)d_in[3];
    const float* box1 = (const float*)d_in[4];
    const float* obj1 = (const float*)d_in[5];
    const float* cls2 = (const float*)d_in[6];
    const float* box2 = (const float*)d_in[7];
    const float* obj2 = (const float*)d_in[8];
    const float* gtb  = (const float*)d_in[9];
    const int*   gtl  = (const int*)d_in[10];
    float* acc = (float*)d_ws;
    float* out = (float*)d_out;

    init_acc_kernel<<<1, 32, 0, stream>>>(acc);

    const int nobj[3] = { in_sizes[2], in_sizes[5], in_sizes[8] };
    const float* objp[3] = { obj0, obj1, obj2 };
    for (int s = 0; s < 3; ++s) {
        int n4 = nobj[s] / 4;
        int blocks = (n4 + 255) / 256;
        if (blocks > 1280) blocks = 1280;
        if (blocks < 1) blocks = 1;
        obj_base_kernel<<<blocks, 256, 0, stream>>>(objp[s], n4, acc + s * 5);
    }

    // B*M = 6400 = 25 blocks of 256 (full waves -> EXEC all-ones at WMMA)
    pos_kernel<<<(BATCH * MBOX) / 256, 256, 0, stream>>>(
        cls0, box0, obj0, cls1, box1, obj1, cls2, box2, obj2, gtb, gtl, acc);

    finalize_kernel<<<1, 1, 0, stream>>>(acc, out);
}